// GatedMLP_69870527971644
// MI455X (gfx1250) — compile-verified
//
#include <hip/hip_runtime.h>

typedef float v2f __attribute__((ext_vector_type(2)));
typedef float v8f __attribute__((ext_vector_type(8)));

#define IN_DIM 5000
#define OUT_DIM 30
#define KSEL 50
#define KPAD 64
#define BATCH 16384
#define TOPK_THREADS 1024
#define WAVES 4
#define BLOCK_T (WAVES * 32)
#define ROWS_PER_BLOCK (WAVES * 16)

// ---------------------------------------------------------------------------
// Kernel 1: top-k selection (stable, ties -> lowest index, like jax top_k),
// writes the k-hot mask, a sorted padded index list, and the gathered
// zero-padded W1 tile Wg[KPAD][32] into workspace.
// ---------------------------------------------------------------------------
__global__ void topk_prep_kernel(const float* __restrict__ logits,
                                 const float* __restrict__ W1,
                                 float* __restrict__ mask_out,
                                 int* __restrict__ idx_ws,
                                 float* __restrict__ wg_ws) {
    __shared__ float vals[IN_DIM];
    __shared__ float rv[TOPK_THREADS];
    __shared__ int   ri[TOPK_THREADS];
    __shared__ int   sel[KSEL];

    const int t  = threadIdx.x;
    const int nt = blockDim.x;

    for (int i = t; i < IN_DIM; i += nt) vals[i] = logits[i];
    __syncthreads();

    for (int k = 0; k < KSEL; ++k) {
        float bv = -__builtin_inff();
        int   bi = IN_DIM;
        for (int i = t; i < IN_DIM; i += nt) {
            float v = vals[i];
            if (v > bv || (v == bv && i < bi)) { bv = v; bi = i; }
        }
        rv[t] = bv; ri[t] = bi;
        __syncthreads();
        for (int s = nt >> 1; s > 0; s >>= 1) {
            if (t < s) {
                float ov = rv[t + s]; int oi = ri[t + s];
                if (ov > rv[t] || (ov == rv[t] && oi < ri[t])) { rv[t] = ov; ri[t] = oi; }
            }
            __syncthreads();
        }
        if (t == 0) { sel[k] = ri[0]; vals[ri[0]] = -__builtin_inff(); }
        __syncthreads();
    }

    // Sort selected indices ascending (mask is set-based; order irrelevant to
    // the math, ascending improves gather locality in the main kernel).
    if (t == 0) {
        for (int i = 1; i < KSEL; ++i) {
            int v = sel[i]; int j = i - 1;
            while (j >= 0 && sel[j] > v) { sel[j + 1] = sel[j]; --j; }
            sel[j + 1] = v;
        }
    }
    __syncthreads();

    // Mask output: one store per address (no cross-thread ordering needed).
    for (int i = t; i < IN_DIM; i += nt) {
        float m = 0.0f;
        for (int k = 0; k < KSEL; ++k) m = (sel[k] == i) ? 1.0f : m;
        mask_out[i] = m;
    }

    // Padded index list (pad -> column 0; its Wg row is zero so it is inert).
    if (t < KPAD) idx_ws[t] = (t < KSEL) ? sel[t] : 0;

    // Gathered, zero-padded W1 tile: Wg[k][j] = W1[sel[k]][j]
    for (int e = t; e < KPAD * 32; e += nt) {
        int kk = e >> 5, j = e & 31;
        wg_ws[e] = (kk < KSEL) ? W1[(size_t)sel[kk] * 32 + j] : 0.0f;
    }
}

// ---------------------------------------------------------------------------
// Kernel 2: fused gather + 3-layer MLP on fp32 WMMA (V_WMMA_F32_16X16X4_F32).
// 4 waves per block, one 16-row tile per wave, 64 rows per block.
// ---------------------------------------------------------------------------
__global__ void mlp_fused_kernel(const float* __restrict__ x,
                                 const int* __restrict__ idx,
                                 const float* __restrict__ wg,
                                 const float* __restrict__ b1,
                                 const float* __restrict__ W2,
                                 const float* __restrict__ b2,
                                 const float* __restrict__ W3,
                                 const float* __restrict__ b3,
                                 float* __restrict__ out) {
    __shared__ int   s_idx[KPAD];
    __shared__ float s_wg[KPAD][32];       // 8 KB
    __shared__ float s_w2[32][16];         // 2 KB
    __shared__ float s_w3[16][32];         // 2 KB (N padded 30->32 with zeros)
    __shared__ float s_b1[32];
    __shared__ float s_b2[16];
    __shared__ float s_b3[32];
    __shared__ float s_h1[WAVES][16][32];  // 8 KB
    __shared__ float s_h2[WAVES][16][16];  // 4 KB

    const int t = threadIdx.x;

    for (int i = t; i < KPAD; i += BLOCK_T) s_idx[i] = idx[i];
    for (int i = t; i < KPAD * 32; i += BLOCK_T) (&s_wg[0][0])[i] = wg[i];
    for (int i = t; i < 32 * 16; i += BLOCK_T) (&s_w2[0][0])[i] = W2[i];
    for (int i = t; i < 16 * 32; i += BLOCK_T) {
        int r = i >> 5, c = i & 31;
        s_w3[r][c] = (c < OUT_DIM) ? W3[r * OUT_DIM + c] : 0.0f;
    }
    if (t < 32) s_b1[t] = b1[t];
    if (t < 16) s_b2[t] = b2[t];
    if (t < 32) s_b3[t] = (t < OUT_DIM) ? b3[t] : 0.0f;
    __syncthreads();

    const int wave = t >> 5;
    const int lane = t & 31;
    const int hi   = lane >> 4;   // 0: lanes 0-15, 1: lanes 16-31
    const int l16  = lane & 15;
    const int rowBase = blockIdx.x * ROWS_PER_BLOCK + wave * 16;
    const float* xrow = x + (size_t)(rowBase + l16) * IN_DIM;

    // ---- GEMM 1: h1[16x32] = relu(Xg[16x64] @ Wg[64x32] + b1) -------------
    v8f acc0 = {};
    v8f acc1 = {};
#pragma unroll
    for (int s = 0; s < KPAD / 4; ++s) {
        const int k0 = 4 * s + 2 * hi;
        // A (16x4 fp32): lanes 0-15 -> K=4s,4s+1 ; lanes 16-31 -> K=4s+2,4s+3
        v2f a;
        a.x = xrow[s_idx[k0]];
        a.y = xrow[s_idx[k0 + 1]];
        // B (4x16 fp32), two N-tiles
        v2f bA, bB;
        bA.x = s_wg[k0][l16];       bA.y = s_wg[k0 + 1][l16];
        bB.x = s_wg[k0][16 + l16];  bB.y = s_wg[k0 + 1][16 + l16];
        acc0 = __builtin_amdgcn_wmma_f32_16x16x4_f32(false, a, false, bA,
                                                     (short)0, acc0, false, false);
        acc1 = __builtin_amdgcn_wmma_f32_16x16x4_f32(false, a, false, bB,
                                                     (short)0, acc1, false, false);
    }
    // C/D layout: VGPR g holds M = g + 8*hi, N = l16 (+16 for acc1)
#pragma unroll
    for (int g = 0; g < 8; ++g) {
        float h0 = fmaxf(acc0[g] + s_b1[l16], 0.0f);
        float h1 = fmaxf(acc1[g] + s_b1[16 + l16], 0.0f);
        s_h1[wave][g + 8 * hi][l16]      = h0;
        s_h1[wave][g + 8 * hi][16 + l16] = h1;
    }
    __syncthreads();

    // ---- GEMM 2: h2[16x16] = relu(h1[16x32] @ W2[32x16] + b2) -------------
    v8f acc2 = {};
#pragma unroll
    for (int s = 0; s < 32 / 4; ++s) {
        const int k0 = 4 * s + 2 * hi;
        v2f a, b;
        a.x = s_h1[wave][l16][k0];
        a.y = s_h1[wave][l16][k0 + 1];
        b.x = s_w2[k0][l16];
        b.y = s_w2[k0 + 1][l16];
        acc2 = __builtin_amdgcn_wmma_f32_16x16x4_f32(false, a, false, b,
                                                     (short)0, acc2, false, false);
    }
#pragma unroll
    for (int g = 0; g < 8; ++g) {
        s_h2[wave][g + 8 * hi][l16] = fmaxf(acc2[g] + s_b2[l16], 0.0f);
    }
    __syncthreads();

    // ---- GEMM 3: out[16x30] = h2[16x16] @ W3[16x30] + b3 ------------------
    v8f acc3 = {};
    v8f acc4 = {};
#pragma unroll
    for (int s = 0; s < 16 / 4; ++s) {
        const int k0 = 4 * s + 2 * hi;
        v2f a, bA, bB;
        a.x = s_h2[wave][l16][k0];
        a.y = s_h2[wave][l16][k0 + 1];
        bA.x = s_w3[k0][l16];       bA.y = s_w3[k0 + 1][l16];
        bB.x = s_w3[k0][16 + l16];  bB.y = s_w3[k0 + 1][16 + l16];
        acc3 = __builtin_amdgcn_wmma_f32_16x16x4_f32(false, a, false, bA,
                                                     (short)0, acc3, false, false);
        acc4 = __builtin_amdgcn_wmma_f32_16x16x4_f32(false, a, false, bB,
                                                     (short)0, acc4, false, false);
    }
#pragma unroll
    for (int g = 0; g < 8; ++g) {
        const int r = rowBase + g + 8 * hi;
        out[(size_t)r * OUT_DIM + l16] = acc3[g] + s_b3[l16];
        if (16 + l16 < OUT_DIM) {
            out[(size_t)r * OUT_DIM + 16 + l16] = acc4[g] + s_b3[16 + l16];
        }
    }
}

// ---------------------------------------------------------------------------
extern "C" void kernel_launch(void* const* d_in, const int* in_sizes, int n_in,
                              void* d_out, int out_size, void* d_ws, size_t ws_size,
                              hipStream_t stream) {
    const float* x      = (const float*)d_in[0];
    const float* logits = (const float*)d_in[1];
    const float* W1     = (const float*)d_in[2];
    const float* b1     = (const float*)d_in[3];
    const float* W2     = (const float*)d_in[4];
    const float* b2     = (const float*)d_in[5];
    const float* W3     = (const float*)d_in[6];
    const float* b3     = (const float*)d_in[7];
    // d_in[8], d_in[9]: epoch / total_epochs (unused in eval path)

    float* out  = (float*)d_out;
    float* mask = out + (size_t)BATCH * OUT_DIM;   // outputs concatenated: (out, mask)

    int*   idx_ws = (int*)d_ws;
    float* wg_ws  = (float*)((char*)d_ws + 256);   // KPAD*32 floats = 8 KB

    topk_prep_kernel<<<1, TOPK_THREADS, 0, stream>>>(logits, W1, mask, idx_ws, wg_ws);

    mlp_fused_kernel<<<BATCH / ROWS_PER_BLOCK, BLOCK_T, 0, stream>>>(
        x, idx_ws, wg_ws, b1, W2, b2, W3, b3, out);
}